// NullMaskedLoRALinear_17626545782941
// MI455X (gfx1250) — compile-verified
//
#include <hip/hip_runtime.h>

typedef __attribute__((ext_vector_type(2))) float v2f;
typedef __attribute__((ext_vector_type(4))) float v4f;
typedef __attribute__((ext_vector_type(8))) float v8f;

#define NROWS 8192
#define KIN   4096
#define NOUT  4096
#define RNK   16

#if defined(__gfx1250__)
#define USE_ASYNC_LDS 1
#else
#define USE_ASYNC_LDS 0
#endif

// ---------------------------------------------------------------- prep: w = B v
__global__ void lora_prep_w(const float* __restrict__ B,
                            const float* __restrict__ v,
                            float* __restrict__ w) {
  __shared__ float red[256];
  const int nidx = blockIdx.x;               // 0..15
  float s = 0.f;
  for (int k = threadIdx.x; k < KIN; k += 256)
    s += B[nidx * KIN + k] * v[k];
  red[threadIdx.x] = s;
  __syncthreads();
  for (int h = 128; h > 0; h >>= 1) {
    if (threadIdx.x < h) red[threadIdx.x] += red[threadIdx.x + h];
    __syncthreads();
  }
  if (threadIdx.x == 0) w[nidx] = red[0];
}

// ---- prep: Wq packs Weff^T so one b128 per lane feeds TWO stage-A B-frags.
// W[k][n] = B[n][k] - w[n]*v[k];  g=k>>3, r=k&7, hi=(r>>1)&1, slot=(r&1)|((r>>2)<<1)
// Wq[((g*2+hi)*16 + n)*4 + slot] = W[k][n]
__global__ void lora_prep_wq(const float* __restrict__ B,
                             const float* __restrict__ v,
                             const float* __restrict__ w,
                             float* __restrict__ Wq) {
  const int idx = blockIdx.x * 256 + threadIdx.x;   // [0, 65536)
  const int nn  = idx >> 12;                        // 0..15
  const int k   = idx & (KIN - 1);                  // 0..4095
  const float val = B[nn * KIN + k] - w[nn] * v[k];
  const int g = k >> 3, r = k & 7;
  const int hi = (r >> 1) & 1;
  const int slot = (r & 1) | ((r >> 2) << 1);
  Wq[(((g * 2 + hi) * 16) + nn) * 4 + slot] = val;
}

// ---- prep: Aq packs A so stage B uses 2 b128 loads per 16-col tile.
// q=k>>2, hi=(k>>1)&1, s=k&1 ; Aq[(j*2+hi)*8 + q*2 + s] = A[j*16 + k]
__global__ void lora_prep_aq(const float* __restrict__ A,
                             float* __restrict__ Aq) {
  const int idx = blockIdx.x * 256 + threadIdx.x;   // [0, 65536)
  const int j = idx >> 4;
  const int k = idx & 15;
  const int q = k >> 2, hi = (k >> 1) & 1, s = k & 1;
  Aq[(j * 2 + hi) * 8 + q * 2 + s] = A[j * 16 + k];
}

// ---------------------------------------------------------------- main kernel
// y[16 rows] = (Xtile @ Weff^T) @ A^T ; block = 16 rows, 8 waves split K / cols
__launch_bounds__(256)
__global__ void lora_main(const float* __restrict__ x,
                          const float* __restrict__ Aq,
                          const float* __restrict__ Wq,
                          float* __restrict__ out) {
  constexpr int CH   = 64;        // K-columns staged per step
  constexpr int XSTR = CH + 4;    // padded LDS row stride (dwords) -> 0 conflicts
  // NOTE: always index xbuf directly (no generic pointer temps) so accesses
  // stay in addrspace(3) and lower to ds_load / async-LDS, never flat_load.
  __shared__ __align__(16) float xbuf[2][8][16 * XSTR];  // double-buffered, per-wave
  __shared__ float twave[8][16 * RNK];                   // per-wave partial T

  const int tid  = threadIdx.x;
  const int lane = tid & 31;
  const int wv   = tid >> 5;
  const int hi   = lane >> 4;      // 0 | 1
  const int n    = lane & 15;      // row for A-frags, col for B/C/D frags
  const int off  = hi << 1;        // K sub-offset 0 | 2
  const int r0   = blockIdx.x * 16;
  const int kw   = wv * 512;       // this wave's K slice base

  // stage one 16x64 x-chunk into LDS buffer b (async DMA on gfx1250)
  auto issue_chunk = [&](int cc, int b) {
    const float* gsrc = x + (size_t)r0 * KIN + kw + cc * CH;
#if USE_ASYNC_LDS
#pragma unroll
    for (int i = 0; i < 8; ++i) {
      const int row = i * 2 + hi;
      const float* gp = gsrc + row * KIN + n * 4;
      unsigned lds_addr = (unsigned)(size_t)&xbuf[b][wv][row * XSTR + n * 4];
      asm volatile("global_load_async_to_lds_b128 %0, %1, off"
                   :: "v"(lds_addr), "v"(gp) : "memory");
    }
#else
#pragma unroll
    for (int i = 0; i < 8; ++i) {
      const int row = i * 2 + hi;
      *(v4f*)&xbuf[b][wv][row * XSTR + n * 4] =
          *(const v4f*)(gsrc + row * KIN + n * 4);
    }
#endif
  };

  // ---------------- stage A: partial T over this wave's K slice ------------
  v8f t0 = {}; v8f t1 = {};
  issue_chunk(0, 0);

  for (int cc = 0; cc < 8; ++cc) {
    if (cc < 7) {
#if USE_ASYNC_LDS
      // WAR: ds_loads from the buffer we are about to overwrite must be done
      asm volatile("s_wait_dscnt 0x0" ::: "memory");
#endif
      issue_chunk(cc + 1, (cc + 1) & 1);
#if USE_ASYNC_LDS
      // async loads retire in order: <=8 outstanding => chunk cc fully landed
      asm volatile("s_wait_asynccnt 0x8" ::: "memory");
#endif
    } else {
#if USE_ASYNC_LDS
      asm volatile("s_wait_asynccnt 0x0" ::: "memory");
#endif
    }

    const int buf   = cc & 1;
    const int gbase = (kw + cc * CH) >> 3;       // k-group base for Wq
    const int lbase = n * XSTR;                  // A-frag row base for this lane
#pragma unroll
    for (int c8 = 0; c8 < 8; ++c8) {
      v4f wq = *(const v4f*)(Wq + (((size_t)(gbase + c8) * 2 + hi) * 16 + n) * 4);
      v2f a0 = *(const v2f*)&xbuf[buf][wv][lbase + c8 * 8 + off];
      v2f b0 = __builtin_shufflevector(wq, wq, 0, 1);
      t0 = __builtin_amdgcn_wmma_f32_16x16x4_f32(false, a0, false, b0,
                                                 (short)0, t0, false, false);
      v2f a1 = *(const v2f*)&xbuf[buf][wv][lbase + c8 * 8 + 4 + off];
      v2f b1 = __builtin_shufflevector(wq, wq, 2, 3);
      t1 = __builtin_amdgcn_wmma_f32_16x16x4_f32(false, a1, false, b1,
                                                 (short)0, t1, false, false);
    }
  }
#pragma unroll
  for (int g = 0; g < 8; ++g)
    twave[wv][(g + hi * 8) * RNK + n] = t0[g] + t1[g];
  __syncthreads();

  // ------- deterministic cross-wave reduction -> 4 stage-B A-fragments -----
  v2f ta[4];
#pragma unroll
  for (int q = 0; q < 4; ++q) {
    float a0 = 0.f, a1 = 0.f;
#pragma unroll
    for (int w8 = 0; w8 < 8; ++w8) {
      a0 += twave[w8][n * RNK + q * 4 + off];
      a1 += twave[w8][n * RNK + q * 4 + off + 1];
    }
    ta[q][0] = a0; ta[q][1] = a1;
  }

  // ---------------- stage B: Y = T @ A^T over this wave's 512 columns ------
  const int c0 = wv * 512;
  for (int oc = 0; oc < 32; ++oc) {
    const int n0 = c0 + oc * 16;
    const float* aq = Aq + ((size_t)(n0 + n) * 2 + hi) * 8;
    v4f q01 = *(const v4f*)(aq);
    v4f q23 = *(const v4f*)(aq + 4);
    v8f y0 = {}; v8f y1 = {};
    y0 = __builtin_amdgcn_wmma_f32_16x16x4_f32(
        false, ta[0], false, __builtin_shufflevector(q01, q01, 0, 1),
        (short)0, y0, false, false);
    y1 = __builtin_amdgcn_wmma_f32_16x16x4_f32(
        false, ta[1], false, __builtin_shufflevector(q01, q01, 2, 3),
        (short)0, y1, false, false);
    y0 = __builtin_amdgcn_wmma_f32_16x16x4_f32(
        false, ta[2], false, __builtin_shufflevector(q23, q23, 0, 1),
        (short)0, y0, false, false);
    y1 = __builtin_amdgcn_wmma_f32_16x16x4_f32(
        false, ta[3], false, __builtin_shufflevector(q23, q23, 2, 3),
        (short)0, y1, false, false);
    float* orow = out + (size_t)(r0 + hi * 8) * NOUT + n0 + n;
#pragma unroll
    for (int g = 0; g < 8; ++g)
      orow[(size_t)g * NOUT] = y0[g] + y1[g];
  }
}

// ---------------------------------------------------------------- launcher
extern "C" void kernel_launch(void* const* d_in, const int* in_sizes, int n_in,
                              void* d_out, int out_size, void* d_ws, size_t ws_size,
                              hipStream_t stream) {
  (void)in_sizes; (void)n_in; (void)out_size; (void)ws_size;
  const float* x = (const float*)d_in[0];   // [8192,4096]
  const float* A = (const float*)d_in[1];   // [4096,16]
  const float* B = (const float*)d_in[2];   // [16,4096]
  const float* v = (const float*)d_in[3];   // [4096]
  float* out = (float*)d_out;               // [8192,4096]

  float* w  = (float*)d_ws;                 // 16 floats
  float* Wq = w + 16;                       // 65536 floats (256 KB)
  float* Aq = Wq + 65536;                   // 65536 floats (256 KB)

  lora_prep_w <<<16,  256, 0, stream>>>(B, v, w);
  lora_prep_wq<<<256, 256, 0, stream>>>(B, v, w, Wq);
  lora_prep_aq<<<256, 256, 0, stream>>>(A, Aq);
  lora_main   <<<NROWS / 16, 256, 0, stream>>>(x, Aq, Wq, out);
}